// SparseChannel2Spatial_14568529068642
// MI455X (gfx1250) — compile-verified
//
#include <hip/hip_runtime.h>
#include <stdint.h>

// Sparse octree subdivision (channel -> spatial), MI455X / gfx1250.
// Pure data-movement op: ~0.5 GB HBM traffic, 0 FLOPs -> ~21 us roofline
// at 23.3 TB/s. No matmul exists in this op, so no WMMA; the CDNA5 lever is
// the async global->LDS DMA path (ASYNCcnt) for the streaming feats reads,
// plus NT cache hints so the read-once/write-once streams don't churn L2.

typedef float v4f __attribute__((ext_vector_type(4)));
typedef int   v4i __attribute__((ext_vector_type(4)));

#define FACTOR 2
#define F3     8      // children per voxel
#define KACT   4      // active children per voxel (static)
#define CCH    64     // input channels
#define CH     8      // output channels per child = CCH / F3
#define VPB    64     // voxels per block
#define TPB    256    // threads per block = VPB * KACT  (8 wave32s)
#define TILE_FLOATS (VPB * CCH)   // 4096 floats = 16 KB LDS tile

__global__ __launch_bounds__(TPB) void subdiv_c2s_kernel(
    const float* __restrict__ feats,
    const int*   __restrict__ coords,
    const int*   __restrict__ sub,
    float*       __restrict__ out_feats,   // [M, 8] float32
    int*         __restrict__ out_coords,  // [M, 4] int32
    int n)                                 // number of voxels
{
  __shared__ float tile[TILE_FLOATS];

  const int t = threadIdx.x;
  const long long vbase = (long long)blockIdx.x * VPB;

  // ---------------------------------------------------------------------
  // Phase 1: stage this block's feats tile (64 voxels x 256B = 16 KB) into
  // LDS with CDNA5 async global->LDS B128 copies (tracked by ASYNCcnt).
  // HBM sees a perfectly sequential, full-cacheline, non-temporal read
  // stream regardless of the sparsity mask. The LDS destination is
  // XOR-swizzled per 32B child chunk (slot j stored at slot j ^ (voxel & 7))
  // so the data-dependent gather in phase 2 is bank-conflict-free.
  //
  // Per-iteration stride is +4096B on BOTH sides (16 voxels x 256B in
  // global; 16 rows x 256B in LDS, swizzle key lv&7 invariant under
  // lv += 16), and the async-load INST_OFFSET is added to both the global
  // and LDS addresses -> one address setup + immediate offsets.
  // ---------------------------------------------------------------------
  if (vbase + VPB <= (long long)n) {
    // Fast path: whole block in range (always taken for N % 64 == 0).
    const int lv    = t >> 4;          // local voxel 0..15 (iteration 0)
    const int sub16 = t & 15;          // 16B sub-chunk within 256B row
    const int j     = sub16 >> 1;      // child slot 0..7
    const int h     = sub16 & 1;       // low/high half of the 32B chunk
    const int js    = j ^ (lv & 7);    // swizzled child slot
    const float* g  = feats + (vbase + lv) * CCH + j * CH + h * 4;
    const unsigned lds_off =
        (unsigned)(uintptr_t)&tile[lv * CCH + js * CH + h * 4];
    asm volatile(
        "global_load_async_to_lds_b128 %0, %1, off th:TH_LOAD_NT\n\t"
        "global_load_async_to_lds_b128 %0, %1, off offset:4096 th:TH_LOAD_NT\n\t"
        "global_load_async_to_lds_b128 %0, %1, off offset:8192 th:TH_LOAD_NT\n\t"
        "global_load_async_to_lds_b128 %0, %1, off offset:12288 th:TH_LOAD_NT"
        :: "v"(lds_off), "v"(g)
        : "memory");
  } else {
    // Cold fallback: per-chunk guarded staging for a partial tail block.
    #pragma unroll
    for (int i = 0; i < 4; ++i) {
      const int ch16  = t + i * TPB;   // 16B chunk id in tile, [0, 1024)
      const int lv    = ch16 >> 4;
      const int sub16 = ch16 & 15;
      const int j     = sub16 >> 1;
      const int h     = sub16 & 1;
      const int js    = j ^ (lv & 7);
      if (vbase + lv < n) {
        const float* g = feats + (vbase + lv) * CCH + j * CH + h * 4;
        const unsigned lds_off =
            (unsigned)(uintptr_t)&tile[lv * CCH + js * CH + h * 4];
        asm volatile("global_load_async_to_lds_b128 %0, %1, off th:TH_LOAD_NT"
                     :: "v"(lds_off), "v"(g)
                     : "memory");
      }
    }
  }
  // Wait for this wave's async DMAs, then barrier so all waves' LDS writes
  // are visible block-wide.
  asm volatile("s_wait_asynccnt 0" ::: "memory");
  __syncthreads();

  // ---------------------------------------------------------------------
  // Phase 2: one thread per output row m = 4*voxel + k.
  // ---------------------------------------------------------------------
  const int lv = t >> 2;          // local voxel
  const int k  = t & 3;           // which active child (rank among set bits)
  const long long vox = vbase + lv;
  if (vox >= n) return;

  // Build 8-bit child mask from the sub row (two b128 loads).
  const v4i* srow = (const v4i*)(sub + vox * F3);
  const v4i s0 = srow[0];
  const v4i s1 = srow[1];
  unsigned mask = (s0.x != 0 ?   1u : 0u) | (s0.y != 0 ?   2u : 0u) |
                  (s0.z != 0 ?   4u : 0u) | (s0.w != 0 ?   8u : 0u) |
                  (s1.x != 0 ?  16u : 0u) | (s1.y != 0 ?  32u : 0u) |
                  (s1.z != 0 ?  64u : 0u) | (s1.w != 0 ? 128u : 0u);

  // Select the k-th set bit (branchless: k <= 3).
  unsigned mm = mask;
  if (k > 0) mm &= mm - 1u;
  if (k > 1) mm &= mm - 1u;
  if (k > 2) mm &= mm - 1u;
  mm = mm ? mm : 1u;                       // defensive; reference has 4 bits
  const int j = __builtin_ctz(mm);         // child slot 0..7

  // new_coords = coords * [1,2,2,2] + octant offset [0, j&1, (j>>1)&1, (j>>2)&1]
  const v4i c = *(const v4i*)(coords + vox * 4);
  v4i nc;
  nc.x = c.x;
  nc.y = c.y * FACTOR + (j & 1);
  nc.z = c.z * FACTOR + ((j >> 1) & 1);
  nc.w = c.w * FACTOR + ((j >> 2) & 1);

  const long long m = vox * KACT + k;      // static output row

  // Write-once streaming outputs: non-temporal (TH=NT) stores keep L2 free.
  __builtin_nontemporal_store(nc, (v4i*)(out_coords + m * 4));

  // Gather the 32B feature chunk for child j from the swizzled LDS tile.
  const int js = j ^ (lv & 7);
  const v4f f0 = *(const v4f*)&tile[lv * CCH + js * CH];
  const v4f f1 = *(const v4f*)&tile[lv * CCH + js * CH + 4];
  __builtin_nontemporal_store(f0, (v4f*)(out_feats + m * CH));
  __builtin_nontemporal_store(f1, (v4f*)(out_feats + m * CH + 4));
}

extern "C" void kernel_launch(void* const* d_in, const int* in_sizes, int n_in,
                              void* d_out, int out_size, void* d_ws, size_t ws_size,
                              hipStream_t stream) {
  // setup_inputs order: feats [N,64] f32, coords [N,4] i32, sub [N,8] i32
  const float* feats  = (const float*)d_in[0];
  const int*   coords = (const int*)d_in[1];
  const int*   sub    = (const int*)d_in[2];

  const int n = in_sizes[0] / CCH;                 // N voxels
  const long long M = (long long)n * KACT;         // output rows

  // d_out = new_feats [M,8] f32 followed by new_coords [M,4] i32 (flat).
  float* out_feats  = (float*)d_out;
  int*   out_coords = (int*)((float*)d_out + M * CH);

  const int grid = (n + VPB - 1) / VPB;            // N=1e6 -> 15625 blocks
  subdiv_c2s_kernel<<<grid, TPB, 0, stream>>>(feats, coords, sub,
                                              out_feats, out_coords, n);
}